// IOUloss3D_12352325943693
// MI455X (gfx1250) — compile-verified
//
#include <hip/hip_runtime.h>

// 3D-IoU loss, memory-bound elementwise kernel for MI455X (gfx1250, wave32).
//
// Per box: load 2x float4 (pred) + 2x float4 (target) -> 64 B, store 1 float.
// 1.5M boxes -> ~102 MB traffic -> ~4.4 us at 23.3 TB/s HBM (fits in 192 MB L2).
// ~90 flops/box => AI ~1.3 flop/byte: strictly bandwidth-bound, no WMMA role.
//
// The reference's 24-triple-product hexahedron volume has all-planar faces,
// so it equals the prismatoid formula exactly:
//   V = (A0 + 4*Amid + A1) / 6,  4*Amid = (w0x+w1x)*(w0y+w1y).

__device__ __forceinline__ float frustum_vol(
    float y1, float x1, float y2, float x2,
    float y1p, float x1p, float y2p, float x2p,
    bool& en)
{
    en = (x1 < x2) & (y1 < y2) & (x1p < x2p) & (y1p < y2p);
    float w0x = x2 - x1;
    float w0y = y2 - y1;
    float w1x = x2p - x1p;
    float w1y = y2p - y1p;
    float a0 = w0x * w0y;
    float a1 = w1x * w1y;
    float am4 = (w0x + w1x) * (w0y + w1y);   // 4 * mid-plane area
    float v = (a0 + am4 + a1) * (1.0f / 6.0f);
    return en ? v : 0.0f;
}

__global__ void __launch_bounds__(256)
iou3d_loss_kernel(const float4* __restrict__ pred,
                  const float4* __restrict__ targ,
                  float* __restrict__ out,
                  int n)
{
    int i = blockIdx.x * blockDim.x + threadIdx.x;
    if (i >= n) return;

    // Row-major [N,8] fp32: each box = two float4s -> global_load_b128 x4.
    float4 pa = pred[2 * i + 0];
    float4 pb = pred[2 * i + 1];
    float4 ta = targ[2 * i + 0];
    float4 tb = targ[2 * i + 1];

    // _corners: c = [xy - wh/2, xy + wh/2, xy' - wh'/2, xy' + wh'/2], clamped >= 0
    float py1 = fmaxf(pa.x - 0.5f * pa.z, 0.0f);
    float px1 = fmaxf(pa.y - 0.5f * pa.w, 0.0f);
    float py2 = fmaxf(pa.x + 0.5f * pa.z, 0.0f);
    float px2 = fmaxf(pa.y + 0.5f * pa.w, 0.0f);
    float py1p = fmaxf(pb.x - 0.5f * pb.z, 0.0f);
    float px1p = fmaxf(pb.y - 0.5f * pb.w, 0.0f);
    float py2p = fmaxf(pb.x + 0.5f * pb.z, 0.0f);
    float px2p = fmaxf(pb.y + 0.5f * pb.w, 0.0f);

    float ty1 = fmaxf(ta.x - 0.5f * ta.z, 0.0f);
    float tx1 = fmaxf(ta.y - 0.5f * ta.w, 0.0f);
    float ty2 = fmaxf(ta.x + 0.5f * ta.z, 0.0f);
    float tx2 = fmaxf(ta.y + 0.5f * ta.w, 0.0f);
    float ty1p = fmaxf(tb.x - 0.5f * tb.z, 0.0f);
    float tx1p = fmaxf(tb.y - 0.5f * tb.w, 0.0f);
    float ty2p = fmaxf(tb.x + 0.5f * tb.z, 0.0f);
    float tx2p = fmaxf(tb.y + 0.5f * tb.w, 0.0f);

    bool en1, en2, en3;
    float pred_v = frustum_vol(py1, px1, py2, px2, py1p, px1p, py2p, px2p, en1);
    float tgt_v  = frustum_vol(ty1, tx1, ty2, tx2, ty1p, tx1p, ty2p, tx2p, en2);

    // Intersection rectangles at z=0 and z=1
    float ix1 = fmaxf(px1, tx1);
    float iy1 = fmaxf(py1, ty1);
    float ix2 = fminf(px2, tx2);
    float iy2 = fminf(py2, ty2);
    float ix1p = fmaxf(px1p, tx1p);
    float iy1p = fmaxf(py1p, ty1p);
    float ix2p = fminf(px2p, tx2p);
    float iy2p = fminf(py2p, ty2p);

    float inter_v = frustum_vol(iy1, ix1, iy2, ix2, iy1p, ix1p, iy2p, ix2p, en3);

    bool en = en1 & en2 & en3;
    inter_v = en ? inter_v : 0.0f;

    float iou = inter_v / (tgt_v + pred_v - inter_v + 1e-16f);
    out[i] = 1.0f - iou * iou;
}

extern "C" void kernel_launch(void* const* d_in, const int* in_sizes, int n_in,
                              void* d_out, int out_size, void* d_ws, size_t ws_size,
                              hipStream_t stream)
{
    const float4* pred = (const float4*)d_in[0];   // [N, 8] fp32
    const float4* targ = (const float4*)d_in[1];   // [N, 8] fp32
    float* out = (float*)d_out;                    // [N] fp32

    int n = in_sizes[0] / 8;                       // number of boxes
    if (n <= 0) return;

    const int block = 256;                          // 8 wave32s per workgroup
    const int grid = (n + block - 1) / block;
    iou3d_loss_kernel<<<grid, block, 0, stream>>>(pred, targ, out, n);
}